// simpleGCN_39788577030710
// MI455X (gfx1250) — compile-verified
//
#include <hip/hip_runtime.h>
#include <math.h>

typedef __attribute__((ext_vector_type(2))) float v2f;
typedef __attribute__((ext_vector_type(8))) float v8f;

#define NFEAT 128
#define NCLS  32

// ---------------------------------------------------------------- utilities

__global__ void zero_f32_kernel(float* __restrict__ p, long n) {
  long i = (long)blockIdx.x * blockDim.x + threadIdx.x;
  if (i < n) p[i] = 0.0f;
}

// deg[row[e]] += 1  (float atomics; exact for counts < 2^24)
__global__ void degree_kernel(const int* __restrict__ row,
                              float* __restrict__ deg, int nedges) {
  int e = blockIdx.x * blockDim.x + threadIdx.x;
  if (e < nedges) atomicAdd(&deg[row[e]], 1.0f);
}

// deg -> deg_inv_sqrt in place
__global__ void rsqrt_kernel(float* __restrict__ deg, int n) {
  int i = blockIdx.x * blockDim.x + threadIdx.x;
  if (i < n) {
    float d = deg[i];
    deg[i] = (d > 0.0f) ? rsqrtf(d) : 0.0f;
  }
}

// ---------------------------------------------------------------- WMMA GEMM
// h[N,32] = x[N,128] @ w[128,32], full f32 via V_WMMA_F32_16X16X4_F32.
// One wave per 16-row tile; two 16x16 accumulators cover the 32 columns.
// Weight staged in LDS pre-swizzled as K-pair float2s so each B fragment is
// one aligned ds_load_b64 (no VALU repacking in the WMMA feed path).
__global__ void gemm_wmma_f32(const float* __restrict__ x,
                              const float* __restrict__ w,
                              float* __restrict__ h, int ntiles) {
  __shared__ v2f lw2[(NFEAT / 2) * NCLS];  // 16 KB
  // lw2[k2*32 + n] = { w[2*k2][n], w[2*k2+1][n] }
  for (int i = threadIdx.x; i < (NFEAT / 2) * NCLS; i += blockDim.x) {
    const int k2 = i >> 5;        // 0..63
    const int n  = i & 31;
    v2f v;
    v.x = w[(2 * k2) * NCLS + n];
    v.y = w[(2 * k2 + 1) * NCLS + n];
    lw2[i] = v;
  }
  __syncthreads();

  const int wave  = threadIdx.x >> 5;
  const int lane  = threadIdx.x & 31;
  const int mtile = blockIdx.x * (blockDim.x >> 5) + wave;
  if (mtile >= ntiles) return;

  const int m0    = mtile * 16;
  const int ml    = lane & 15;        // row within tile (A) / col within tile (B)
  const int khalf = (lane >> 4) * 2;  // lanes 16-31 hold K+2,K+3 (always even)

  const float* xrow = x + (size_t)(m0 + ml) * NFEAT;

  v8f acc0 = {};  // columns 0..15
  v8f acc1 = {};  // columns 16..31

  for (int k = 0; k < NFEAT; k += 4) {
    const int kk = k + khalf;            // even
    // A fragment: contiguous float2 from this lane's x row
    v2f a = *(const v2f*)(xrow + kk);
    // B fragments: single 64-bit LDS loads (pre-swizzled layout)
    const int kbase = (kk >> 1) * NCLS;  // (kk/2)*32
    v2f b0 = lw2[kbase + ml];
    v2f b1 = lw2[kbase + 16 + ml];
    // (neg_a, A, neg_b, B, c_mod, C, reuse_a, reuse_b)
    acc0 = __builtin_amdgcn_wmma_f32_16x16x4_f32(false, a, false, b0,
                                                 (short)0, acc0, false, false);
    acc1 = __builtin_amdgcn_wmma_f32_16x16x4_f32(false, a, false, b1,
                                                 (short)0, acc1, false, false);
  }

  // C/D layout: VGPR r, lanes 0-15 -> (M=r, N=lane); lanes 16-31 -> (M=8+r)
  const int nl = lane & 15;
  const int mb = (lane >> 4) * 8;
  float* hp = h + (size_t)(m0 + mb) * NCLS + nl;
#pragma unroll
  for (int r = 0; r < 8; ++r) {
    hp[(size_t)r * NCLS]      = acc0[r];
    hp[(size_t)r * NCLS + 16] = acc1[r];
  }
}

// ---------------------------------------------------------------- propagation
// One wave per edge; lane = channel. Gather h[row] (coalesced 128B),
// scale by dis[row]*dis[col], scatter-add into hout[col] (coalesced atomics).
__global__ void edge_prop_kernel(const int* __restrict__ row,
                                 const int* __restrict__ col,
                                 const float* __restrict__ dis,
                                 const float* __restrict__ hin,
                                 float* __restrict__ hout, int nedges) {
  const long gid = (long)blockIdx.x * blockDim.x + threadIdx.x;
  const int  e   = (int)(gid >> 5);
  const int  c   = (int)(gid & 31);
  if (e >= nedges) return;
  const int r  = row[e];    // uniform across the wave -> scalarized
  const int cc = col[e];
  const float nrm = dis[r] * dis[cc];
  const float v   = hin[(size_t)r * NCLS + c] * nrm;
  atomicAdd(&hout[(size_t)cc * NCLS + c], v);
}

// ---------------------------------------------------------------- riemann agg
// One wave per node: s = sum_c h^2 * |w|; h /= (sqrt(s+0.01)+0.01)
__global__ void riemann_kernel(float* __restrict__ h,
                               const float* __restrict__ wnorm, int nnodes) {
  const long gid = (long)blockIdx.x * blockDim.x + threadIdx.x;
  const int  n   = (int)(gid >> 5);
  const int  c   = (int)(gid & 31);
  if (n >= nnodes) return;
  const float v = h[(size_t)n * NCLS + c];
  float t = v * v * fabsf(wnorm[c]);
#pragma unroll
  for (int off = 16; off > 0; off >>= 1) t += __shfl_xor(t, off, 32);
  h[(size_t)n * NCLS + c] = v / (sqrtf(t + 0.01f) + 0.01f);
}

// ---------------------------------------------------------------- launcher

extern "C" void kernel_launch(void* const* d_in, const int* in_sizes, int n_in,
                              void* d_out, int out_size, void* d_ws, size_t ws_size,
                              hipStream_t stream) {
  const float* x      = (const float*)d_in[0];
  const int*   eidx   = (const int*)d_in[1];
  const float* weight = (const float*)d_in[2];
  const float* wnorm  = (const float*)d_in[3];

  const int nnodes = in_sizes[0] / NFEAT;
  const int nedges = in_sizes[1] / 2;
  const int* row = eidx;
  const int* col = eidx + nedges;

  // workspace: dis[N] | hA[32N] | hB[32N]
  float* dis = (float*)d_ws;
  float* hA  = dis + nnodes;
  float* hB  = hA + (size_t)nnodes * NCLS;
  float* out = (float*)d_out;

  const int BS = 256;
  const long nNC   = (long)nnodes * NCLS;
  const int  gNode = (nnodes + BS - 1) / BS;
  const int  gNC   = (int)((nNC + BS - 1) / BS);
  const int  gEdge = (nedges + BS - 1) / BS;
  const int  gEC   = (int)(((long)nedges * 32 + BS - 1) / BS);

  // degree -> deg_inv_sqrt (in place)
  zero_f32_kernel<<<gNode, BS, 0, stream>>>(dis, nnodes);
  degree_kernel<<<gEdge, BS, 0, stream>>>(row, dis, nedges);
  rsqrt_kernel<<<gNode, BS, 0, stream>>>(dis, nnodes);

  // hA = x @ weight  (WMMA f32)
  const int ntiles = (nnodes + 15) / 16;            // 6250 for N=100000
  const int wavesPerBlk = BS / 32;
  const int gGemm = (ntiles + wavesPerBlk - 1) / wavesPerBlk;
  gemm_wmma_f32<<<gGemm, BS, 0, stream>>>(x, weight, hA, ntiles);

  // layer 1: hA -> hB
  zero_f32_kernel<<<gNC, BS, 0, stream>>>(hB, nNC);
  edge_prop_kernel<<<gEC, BS, 0, stream>>>(row, col, dis, hA, hB, nedges);
  riemann_kernel<<<gNC, BS, 0, stream>>>(hB, wnorm, nnodes);

  // layer 2: hB -> d_out
  zero_f32_kernel<<<gNC, BS, 0, stream>>>(out, nNC);
  edge_prop_kernel<<<gEC, BS, 0, stream>>>(row, col, dis, hB, out, nedges);
  riemann_kernel<<<gNC, BS, 0, stream>>>(out, wnorm, nnodes);
}